// CircuitGnn_15187004358978
// MI455X (gfx1250) — compile-verified
//
#include <hip/hip_runtime.h>
#include <math.h>

#define N_NODES  100000
#define N_EDGES  300000
#define FEAT     256
#define N_GRAPHS 500
#define N_LAYERS 6
#define KTOT     512      // fused K: [mean | x]
#define MTILE    32       // rows per block; 100000 = 32*3125 exactly -> no tail
#define LDSA     260      // LDS row stride (floats) = 256 data + 4 pad (TDM pad) -> conflict-free

typedef __attribute__((ext_vector_type(2))) float v2f;
typedef __attribute__((ext_vector_type(8))) float v8f;
typedef unsigned int tdm_v4u __attribute__((ext_vector_type(4)));
typedef int          tdm_v8i __attribute__((ext_vector_type(8)));
typedef int          tdm_v4i __attribute__((ext_vector_type(4)));

// ---------------- TDM: async copy of a 2D fp32 tile (FEAT x MTILE) global->LDS ----------------
// D# per CDNA5 ISA: group0 = {count, lds_addr, global_addr, type=2},
// group1 = {data_size=4B, pad_enable, pad_interval=256 DW, pad_amount=4 DW, dims/strides/tile}.
// This toolchain exposes the 6-arg builtin: (v4u g0, v8i g1, v4i g2, v4i g3, v8i g4, i32 cpol).
__device__ __forceinline__ void tdm_load_tile(const float* gsrc, void* lds_dst) {
    unsigned long long ga = (unsigned long long)gsrc;
    unsigned lds_off = (unsigned)(unsigned long long)lds_dst;  // LDS aperture: addr[31:0] = LDS byte offset
    tdm_v4u g0;
    g0.x = 1u;                                                  // count=1 valid descriptor
    g0.y = lds_off;                                             // lds_addr
    g0.z = (unsigned)(ga & 0xFFFFFFFFu);                        // global_addr[31:0]
    g0.w = (unsigned)((ga >> 32) & 0x1FFFFFFu) | (2u << 30);    // global_addr[56:32] | type=2
    tdm_v8i g1;
    g1[0] = (int)((2u << 16) | (1u << 20) | (7u << 22) | (3u << 25)); // data_size=4B, pad_en, ivl=256DW, amt=4DW
    g1[1] = (int)(FEAT << 16);                                  // tensor_dim0[15:0] = 256
    g1[2] = (int)((N_NODES & 0xFFFFu) << 16);                   // tensor_dim0 hi=0 | tensor_dim1 lo16
    g1[3] = (int)((N_NODES >> 16) | ((unsigned)FEAT << 16));    // tensor_dim1 hi | tile_dim0 = 256
    g1[4] = MTILE;                                              // tile_dim1 = 32
    g1[5] = FEAT;                                               // tensor_dim0_stride = 256 (low 32)
    g1[6] = 0;
    g1[7] = 0;
    tdm_v4i z4 = {0, 0, 0, 0};
    tdm_v8i z8 = {0, 0, 0, 0, 0, 0, 0, 0};
    __builtin_amdgcn_tensor_load_to_lds(g0, g1, z4, z4, z8, 0);
}

// ---------------- degree (once) ----------------
__global__ void deg_kernel(const long long* __restrict__ dst, float* __restrict__ deg) {
    int e = blockIdx.x * blockDim.x + threadIdx.x;
    if (e < N_EDGES) atomicAdd(&deg[(int)dst[e]], 1.0f);
}

// ---------------- edge scatter: msg[dst] += x[src] (wave per edge) ----------------
__global__ __launch_bounds__(256) void scatter_kernel(const float* __restrict__ x,
                                                      const long long* __restrict__ src,
                                                      const long long* __restrict__ dst,
                                                      float* __restrict__ msg) {
    int tid  = blockIdx.x * blockDim.x + threadIdx.x;
    int e    = tid >> 5;
    int lane = tid & 31;
    if (e >= N_EDGES) return;
    int s = (int)src[e];
    int d = (int)dst[e];
    const float4* xs = (const float4*)(x + (size_t)s * FEAT);
    float*        md = msg + (size_t)d * FEAT;
#pragma unroll
    for (int i = 0; i < 2; ++i) {
        float4 v = xs[lane * 2 + i];
        int c = (lane * 2 + i) * 4;
        atomicAdd(md + c + 0, v.x);
        atomicAdd(md + c + 1, v.y);
        atomicAdd(md + c + 2, v.z);
        atomicAdd(md + c + 3, v.w);
    }
}

// ---------------- build Bcat[512][256]: Bcat[k][n] = k<256 ? Wl[n][k] : Wr[n][k-256] ----------------
__global__ void buildB_kernel(const float* __restrict__ Wl, const float* __restrict__ Wr,
                              float* __restrict__ Bcat) {
    int idx = blockIdx.x * blockDim.x + threadIdx.x;
    if (idx >= KTOT * FEAT) return;
    int k = idx / FEAT, n = idx % FEAT;
    Bcat[idx] = (k < FEAT) ? Wl[n * FEAT + k] : Wr[n * FEAT + (k - FEAT)];
}

// ---------------- fused SAGE layer: out = relu([msg/deg | x] @ Bcat + bl) + x  (fp32 WMMA) ----------------
__global__ __launch_bounds__(256) void sage_gemm_kernel(const float* __restrict__ x,
                                                        const float* __restrict__ msg,
                                                        const float* __restrict__ deg,
                                                        const float* __restrict__ Bcat,
                                                        const float* __restrict__ bias,
                                                        float* __restrict__ out) {
    __shared__ float As1[MTILE * LDSA];   // raw msg tile (scaled at fragment read)
    __shared__ float As2[MTILE * LDSA];   // x tile (also residual source)
    __shared__ float sInv[MTILE];
    const int row0 = blockIdx.x * MTILE;
    const int tid  = threadIdx.x;

    if (tid == 0) {
        tdm_load_tile(msg + (size_t)row0 * FEAT, As1);   // TENSORcnt += 1
        tdm_load_tile(x   + (size_t)row0 * FEAT, As2);   // TENSORcnt += 1
        __builtin_amdgcn_s_wait_tensorcnt(0);
    }
    if (tid < MTILE) {
        sInv[tid] = 1.0f / fmaxf(deg[row0 + tid], 1.0f);
    }
    __syncthreads();

    const int wave  = tid >> 5;
    const int lane  = tid & 31;
    const int rA    = lane & 15;         // A row within 16-row tile / D col index
    const int khalf = (lane >> 4) * 2;   // per documented 16x4 fp32 A/B layout
    const int nA    = wave * 32 + rA;
    const int nB    = nA + 16;

    const float inv0 = sInv[rA];
    const float inv1 = sInv[16 + rA];

    v8f acc00 = {}, acc01 = {}, acc10 = {}, acc11 = {};
    const float* A1r0 = &As1[rA * LDSA + khalf];
    const float* A1r1 = &As1[(16 + rA) * LDSA + khalf];
    const float* A2r0 = &As2[rA * LDSA + khalf];
    const float* A2r1 = &As2[(16 + rA) * LDSA + khalf];

    // half 1: mean part (A = msg * 1/deg), B rows 0..255
    for (int k0 = 0; k0 < FEAT; k0 += 4) {
        int kb = k0 + khalf;
        v2f b0, b1;
        b0.x = Bcat[(size_t)kb * FEAT + nA];       b0.y = Bcat[(size_t)(kb + 1) * FEAT + nA];
        b1.x = Bcat[(size_t)kb * FEAT + nB];       b1.y = Bcat[(size_t)(kb + 1) * FEAT + nB];
        v2f a0; a0.x = A1r0[k0] * inv0; a0.y = A1r0[k0 + 1] * inv0;
        v2f a1; a1.x = A1r1[k0] * inv1; a1.y = A1r1[k0 + 1] * inv1;
        acc00 = __builtin_amdgcn_wmma_f32_16x16x4_f32(false, a0, false, b0, (short)0, acc00, false, false);
        acc01 = __builtin_amdgcn_wmma_f32_16x16x4_f32(false, a0, false, b1, (short)0, acc01, false, false);
        acc10 = __builtin_amdgcn_wmma_f32_16x16x4_f32(false, a1, false, b0, (short)0, acc10, false, false);
        acc11 = __builtin_amdgcn_wmma_f32_16x16x4_f32(false, a1, false, b1, (short)0, acc11, false, false);
    }
    // half 2: root part (A = x), B rows 256..511
    const float* Bx = Bcat + (size_t)FEAT * FEAT;
    for (int k0 = 0; k0 < FEAT; k0 += 4) {
        int kb = k0 + khalf;
        v2f b0, b1;
        b0.x = Bx[(size_t)kb * FEAT + nA];         b0.y = Bx[(size_t)(kb + 1) * FEAT + nA];
        b1.x = Bx[(size_t)kb * FEAT + nB];         b1.y = Bx[(size_t)(kb + 1) * FEAT + nB];
        v2f a0; a0.x = A2r0[k0]; a0.y = A2r0[k0 + 1];
        v2f a1; a1.x = A2r1[k0]; a1.y = A2r1[k0 + 1];
        acc00 = __builtin_amdgcn_wmma_f32_16x16x4_f32(false, a0, false, b0, (short)0, acc00, false, false);
        acc01 = __builtin_amdgcn_wmma_f32_16x16x4_f32(false, a0, false, b1, (short)0, acc01, false, false);
        acc10 = __builtin_amdgcn_wmma_f32_16x16x4_f32(false, a1, false, b0, (short)0, acc10, false, false);
        acc11 = __builtin_amdgcn_wmma_f32_16x16x4_f32(false, a1, false, b1, (short)0, acc11, false, false);
    }

    // epilogue: D element r, lane L -> row = r + 8*(L/16) (+16 per row-tile), col = L%16
    const int   rBase = (lane >> 4) * 8;
    const float bA = bias[nA];
    const float bB = bias[nB];
#pragma unroll
    for (int r = 0; r < 8; ++r) {
        int rr0 = rBase + r;          // row-tile 0
        int rr1 = 16 + rBase + r;     // row-tile 1
        int n0  = row0 + rr0;
        int n1  = row0 + rr1;         // always < N_NODES (32 | 100000)
        float o00 = fmaxf(acc00[r] + bA, 0.0f) + As2[rr0 * LDSA + nA];
        float o01 = fmaxf(acc01[r] + bB, 0.0f) + As2[rr0 * LDSA + nB];
        float o10 = fmaxf(acc10[r] + bA, 0.0f) + As2[rr1 * LDSA + nA];
        float o11 = fmaxf(acc11[r] + bB, 0.0f) + As2[rr1 * LDSA + nB];
        out[(size_t)n0 * FEAT + nA] = o00;
        out[(size_t)n0 * FEAT + nB] = o01;
        out[(size_t)n1 * FEAT + nA] = o10;
        out[(size_t)n1 * FEAT + nB] = o11;
    }
}

// ---------------- per-graph segment bounds ----------------
__global__ void bounds_init(int* __restrict__ start, int* __restrict__ end) {
    int g = blockIdx.x * blockDim.x + threadIdx.x;
    if (g < N_GRAPHS) { start[g] = 0x7fffffff; end[g] = -1; }
}
__global__ void bounds_kernel(const long long* __restrict__ batch,
                              int* __restrict__ start, int* __restrict__ end) {
    int i = blockIdx.x * blockDim.x + threadIdx.x;
    if (i < N_NODES) {
        int g = (int)batch[i];
        atomicMin(&start[g], i);
        atomicMax(&end[g], i);
    }
}

// ---------------- softmax aggregation: block per graph, thread per feature ----------------
__global__ __launch_bounds__(256) void aggr_kernel(const float* __restrict__ x,
                                                   const int* __restrict__ start,
                                                   const int* __restrict__ end,
                                                   const float* __restrict__ tptr,
                                                   float* __restrict__ out) {
    int g = blockIdx.x;
    int f = threadIdx.x;
    int s = start[g], e = end[g];
    if (s > e) { out[(size_t)g * FEAT + f] = 0.0f; return; }
    float t = tptr[0];
    float m = -INFINITY;
    for (int i = s; i <= e; ++i) m = fmaxf(m, t * x[(size_t)i * FEAT + f]);
    float se = 0.0f, sex = 0.0f;
    for (int i = s; i <= e; ++i) {
        float xv = x[(size_t)i * FEAT + f];
        float ee = expf(t * xv - m);
        se  += ee;
        sex += ee * xv;
    }
    out[(size_t)g * FEAT + f] = sex / fmaxf(se, 1e-16f);
}

extern "C" void kernel_launch(void* const* d_in, const int* in_sizes, int n_in,
                              void* d_out, int out_size, void* d_ws, size_t ws_size,
                              hipStream_t stream) {
    const float*     x_in  = (const float*)d_in[0];
    const long long* eidx  = (const long long*)d_in[1];   // [2, N_EDGES] int64
    const long long* batch = (const long long*)d_in[2];
    const float*     Wl    = (const float*)d_in[3];       // [6,256,256]
    const float*     bl    = (const float*)d_in[4];       // [6,256]
    const float*     Wr    = (const float*)d_in[5];       // [6,256,256]
    const float*     tpar  = (const float*)d_in[6];

    const long long* src = eidx;
    const long long* dst = eidx + N_EDGES;

    // workspace layout (256B aligned)
    const size_t XBYTES = (size_t)N_NODES * FEAT * sizeof(float);   // 102.4 MB
    char* ws = (char*)d_ws;
    float* bufA = (float*)ws;                       ws += (XBYTES + 255) & ~(size_t)255;
    float* bufB = (float*)ws;                       ws += (XBYTES + 255) & ~(size_t)255;
    float* deg  = (float*)ws;                       ws += ((size_t)N_NODES * 4 + 255) & ~(size_t)255;
    float* Bcat = (float*)ws;                       ws += ((size_t)KTOT * FEAT * 4 + 255) & ~(size_t)255;
    int*   gs   = (int*)ws;                         ws += ((size_t)N_GRAPHS * 4 + 255) & ~(size_t)255;
    int*   ge   = (int*)ws;                         ws += ((size_t)N_GRAPHS * 4 + 255) & ~(size_t)255;

    // degree once
    (void)hipMemsetAsync(deg, 0, (size_t)N_NODES * 4, stream);
    deg_kernel<<<(N_EDGES + 255) / 256, 256, 0, stream>>>(dst, deg);

    float* bufs[2] = { bufA, bufB };
    const float* cur = x_in;
    for (int l = 0; l < N_LAYERS; ++l) {
        float* msg = bufs[l & 1];
        (void)hipMemsetAsync(msg, 0, XBYTES, stream);
        scatter_kernel<<<(N_EDGES * 32 + 255) / 256, 256, 0, stream>>>(cur, src, dst, msg);
        buildB_kernel<<<(KTOT * FEAT + 255) / 256, 256, 0, stream>>>(
            Wl + (size_t)l * FEAT * FEAT, Wr + (size_t)l * FEAT * FEAT, Bcat);
        sage_gemm_kernel<<<N_NODES / MTILE, 256, 0, stream>>>(
            cur, msg, deg, Bcat, bl + (size_t)l * FEAT, msg /* in-place over msg */);
        cur = msg;
    }

    // final aggregation
    bounds_init<<<(N_GRAPHS + 255) / 256, 256, 0, stream>>>(gs, ge);
    bounds_kernel<<<(N_NODES + 255) / 256, 256, 0, stream>>>(batch, gs, ge);
    aggr_kernel<<<N_GRAPHS, 256, 0, stream>>>(cur, gs, ge, tpar, (float*)d_out);
}